// Upsample_GNN_15290083573885
// MI455X (gfx1250) — compile-verified
//
#include <hip/hip_runtime.h>

#define BATCH 4
#define NPTS  4096
#define CCH   64
#define HEADS 4
#define KNN   16
#define DX    67
#define XPAD  68
#define NN    (BATCH * NPTS)   // 16384
#define CHUNK 512
#define NEG_SLOPE 0.2f

typedef __attribute__((ext_vector_type(2))) float v2f;
typedef __attribute__((ext_vector_type(8))) float v8f;

__device__ __forceinline__ float eluf(float v) { return v > 0.f ? v : (expf(v) - 1.f); }

// ---------------------------------------------------------------------------
// Pack [features ; coords] into row-major padded node table + squared norms.
__global__ void build_x_kernel(const float* __restrict__ pf, const float* __restrict__ pt,
                               float* __restrict__ xbuf, float* __restrict__ x2b) {
    int t = blockIdx.x * blockDim.x + threadIdx.x;
    if (t >= NN) return;
    int b = t / NPTS, n = t % NPTS;
    float* xr = xbuf + (size_t)t * XPAD;
    float s = 0.f;
    for (int c = 0; c < CCH; ++c) {
        float v = pf[((size_t)b * CCH + c) * NPTS + n];
        xr[c] = v; s += v * v;
    }
    for (int j = 0; j < 3; ++j) {
        float v = pt[((size_t)b * 3 + j) * NPTS + n];
        xr[CCH + j] = v; s += v * v;
    }
    xr[DX] = 0.f;              // pad column so K-loop can read k=64..67
    x2b[t] = s;
}

// Squared row norms of an arbitrary strided row-major matrix (for f1).
__global__ void sqnorm_kernel(const float* __restrict__ X, int stride, int D,
                              float* __restrict__ x2b) {
    int t = blockIdx.x * blockDim.x + threadIdx.x;
    if (t >= NN) return;
    const float* xr = X + (size_t)t * stride;
    float s = 0.f;
    for (int c = 0; c < D; ++c) { float v = xr[c]; s += v * v; }
    x2b[t] = s;
}

// ---------------------------------------------------------------------------
// Pairwise-distance block via V_WMMA_F32_16X16X4_F32 with 2x4 register
// blocking: one wave computes a 32x64 block of d(i,j) = |xi|^2+|xj|^2-2<xi,xj>.
// Per k-step: 2 A-fragment loads + 4 B-fragment loads feed 8 WMMAs.
__global__ void __launch_bounds__(128)
dist_tile_kernel(const float* __restrict__ X, const float* __restrict__ x2,
                 float* __restrict__ dist, int b, int colBase, int xStride, int nK4) {
    int wave = threadIdx.x >> 5;
    int lane = threadIdx.x & 31;
    int tile = blockIdx.x * 4 + wave;
    const int nCB = CHUNK / 64;                      // 8 column blocks per chunk
    int rb  = tile / nCB;                            // 32-row block index
    int cbt = tile % nCB;                            // 64-col block index
    int rowBase = rb * 32;
    int colBlk  = cbt * 64;
    int m    = lane & 15;
    int half = lane >> 4;

    // A fragments: rows rowBase+m and rowBase+16+m (float2/lane, K=2*half+vgpr)
    const float* Ar0 = X + (size_t)(b * NPTS + rowBase + m) * xStride + 2 * half;
    const float* Ar1 = Ar0 + (size_t)16 * xStride;
    // B fragments: cols colBase+colBlk + 16*s + m
    const float* Br[4];
#pragma unroll
    for (int s = 0; s < 4; ++s)
        Br[s] = X + (size_t)(b * NPTS + colBase + colBlk + 16 * s + m) * xStride + 2 * half;

    __builtin_prefetch(Ar0, 0, 3);
    __builtin_prefetch(Ar1, 0, 3);
#pragma unroll
    for (int s = 0; s < 4; ++s) __builtin_prefetch(Br[s], 0, 3);

    v8f acc[2][4];
#pragma unroll
    for (int r = 0; r < 2; ++r)
#pragma unroll
        for (int s = 0; s < 4; ++s) acc[r][s] = (v8f){};

    for (int ks = 0; ks < nK4; ++ks) {
        v2f a0 = *(const v2f*)(Ar0 + ks * 4);
        v2f a1 = *(const v2f*)(Ar1 + ks * 4);
        v2f bf[4];
#pragma unroll
        for (int s = 0; s < 4; ++s) bf[s] = *(const v2f*)(Br[s] + ks * 4);
#pragma unroll
        for (int s = 0; s < 4; ++s) {
            acc[0][s] = __builtin_amdgcn_wmma_f32_16x16x4_f32(false, a0, false, bf[s],
                                                              (short)0, acc[0][s], false, false);
            acc[1][s] = __builtin_amdgcn_wmma_f32_16x16x4_f32(false, a1, false, bf[s],
                                                              (short)0, acc[1][s], false, false);
        }
    }

    // Result layout: lane -> column n = lane&15, vgpr v -> row v + 8*(lane>=16).
    int nIdx = lane & 15;
#pragma unroll
    for (int s = 0; s < 4; ++s) {
        int colL   = colBlk + 16 * s + nIdx;         // chunk-local column
        int colAbs = colBase + colL;                 // batch-local column
        float x2c = x2[b * NPTS + colAbs];
#pragma unroll
        for (int r = 0; r < 2; ++r) {
#pragma unroll
            for (int v = 0; v < 8; ++v) {
                int rowL = rowBase + 16 * r + v + 8 * half;
                float d = x2[b * NPTS + rowL] + x2c - 2.f * acc[r][s][v];
                if (rowL == colAbs) d = 3.0e38f;     // exclude self (loop=False)
                dist[(size_t)rowL * CHUNK + colL] = d;
            }
        }
    }
}

// ---------------------------------------------------------------------------
// Streaming top-K=16 smallest per row: register-resident sorted list with a
// fully-unrolled branchless insertion chain (constant indices -> no spills).
__global__ void knn_select_kernel(const float* __restrict__ dist,
                                  float* __restrict__ topd, int* __restrict__ topi,
                                  int b, int colBase, int first) {
    int r = blockIdx.x * blockDim.x + threadIdx.x;
    if (r >= NPTS) return;
    int g = b * NPTS + r;

    float bd[KNN]; int bi[KNN];
    if (first) {
#pragma unroll
        for (int s = 0; s < KNN; ++s) { bd[s] = 3.4e38f; bi[s] = 0; }
    } else {
#pragma unroll
        for (int s = 0; s < KNN; ++s) { bd[s] = topd[(size_t)g * KNN + s]; bi[s] = topi[(size_t)g * KNN + s]; }
    }

    const float* dr = dist + (size_t)r * CHUNK;
    for (int j = 0; j < CHUNK; ++j) {
        float nd = dr[j];
        if (nd < bd[KNN - 1]) {
            int ni = b * NPTS + colBase + j;         // globalized neighbor index
#pragma unroll
            for (int s = 0; s < KNN; ++s) {
                bool sw = nd < bd[s];
                float td = bd[s]; int ti = bi[s];
                bd[s] = sw ? nd : td;  bi[s] = sw ? ni : ti;
                nd    = sw ? td : nd;  ni    = sw ? ti : ni;
            }
        }
    }
#pragma unroll
    for (int s = 0; s < KNN; ++s) { topd[(size_t)g * KNN + s] = bd[s]; topi[(size_t)g * KNN + s] = bi[s]; }
}

// ---------------------------------------------------------------------------
// out[M x 64] = X[M x D] @ W[D x 64] via f32 WMMA, 1x4 register blocking:
// one wave computes 16 rows x all 64 cols; A fragment loaded once per k-step.
// X rows padded to xStride >= 4*nK4 (zeros beyond D), W guarded on k < D.
__global__ void __launch_bounds__(128)
matmul_wmma_kernel(const float* __restrict__ X, int xStride,
                   const float* __restrict__ W, int D,
                   float* __restrict__ out, int outStride, int nK4) {
    int wave = threadIdx.x >> 5;
    int lane = threadIdx.x & 31;
    int rowTile = blockIdx.x * 4 + wave;             // 16-row tile index
    int m    = lane & 15;
    int half = lane >> 4;

    const float* Arow = X + (size_t)(rowTile * 16 + m) * xStride + 2 * half;
    __builtin_prefetch(Arow, 0, 3);

    v8f acc[4];
#pragma unroll
    for (int s = 0; s < 4; ++s) acc[s] = (v8f){};

    for (int ks = 0; ks < nK4; ++ks) {
        v2f a = *(const v2f*)(Arow + ks * 4);
        int k0 = ks * 4 + 2 * half;
#pragma unroll
        for (int s = 0; s < 4; ++s) {
            int col = 16 * s + m;
            v2f bv;
            bv.x = (k0     < D) ? W[(size_t)k0 * CCH + col]       : 0.f;
            bv.y = (k0 + 1 < D) ? W[(size_t)(k0 + 1) * CCH + col] : 0.f;
            acc[s] = __builtin_amdgcn_wmma_f32_16x16x4_f32(false, a, false, bv,
                                                           (short)0, acc[s], false, false);
        }
    }

    int nIdx = lane & 15;
#pragma unroll
    for (int s = 0; s < 4; ++s)
#pragma unroll
        for (int v = 0; v < 8; ++v) {
            int row = rowTile * 16 + v + 8 * half;
            out[(size_t)row * outStride + 16 * s + nIdx] = acc[s][v];
        }
}

// ---------------------------------------------------------------------------
// a-coefficients: coeff[node][h] = sum_c h[node][h*16+c] * a[h][c]
__global__ void attn_coeff_kernel(const float* __restrict__ hb, const float* __restrict__ av,
                                  float* __restrict__ outv) {
    int t = blockIdx.x * blockDim.x + threadIdx.x;
    if (t >= NN * HEADS) return;
    int node = t / HEADS, hh = t % HEADS;
    const float* hr = hb + (size_t)node * CCH + hh * 16;
    const float* ar = av + hh * 16;
    float s = 0.f;
#pragma unroll
    for (int c = 0; c < 16; ++c) s += hr[c] * ar[c];
    outv[t] = s;
}

// GAT attention + aggregation for one node per block (64 threads = channels).
__global__ void __launch_bounds__(64)
gat_agg_kernel(const int* __restrict__ topi, const float* __restrict__ hb,
               const float* __restrict__ asrc, const float* __restrict__ adst,
               const float* __restrict__ bias, float* __restrict__ fcat, int colOff) {
    __shared__ int   nb[KNN];
    __shared__ float ew[HEADS * KNN];
    int node = blockIdx.x;
    int t = threadIdx.x;

    if (t < KNN) nb[t] = topi[(size_t)node * KNN + t];
    __syncthreads();

    {   // e[k][h] = leaky_relu(asrc[nbr] + adst[node])
        int k = t & 15, hh = t >> 4;
        float e = asrc[nb[k] * HEADS + hh] + adst[node * HEADS + hh];
        e = e > 0.f ? e : NEG_SLOPE * e;
        ew[hh * KNN + k] = e;
    }
    __syncthreads();

    if (t < HEADS) {                                 // softmax over K neighbors
        float mx = -3.4e38f;
        for (int k = 0; k < KNN; ++k) mx = fmaxf(mx, ew[t * KNN + k]);
        float sum = 0.f;
        for (int k = 0; k < KNN; ++k) { float v = expf(ew[t * KNN + k] - mx); ew[t * KNN + k] = v; sum += v; }
        float inv = 1.f / sum;
        for (int k = 0; k < KNN; ++k) ew[t * KNN + k] *= inv;
    }
    __syncthreads();

    int hh = t >> 4;                                 // t = output channel
    float acc = 0.f;
    for (int k = 0; k < KNN; ++k)
        acc += ew[hh * KNN + k] * hb[(size_t)nb[k] * CCH + t];
    fcat[(size_t)node * (2 * CCH) + colOff + t] = eluf(acc + bias[t]);
}

// Final bias + ELU + transpose to [B, C, N].
__global__ void final_out_kernel(const float* __restrict__ fo, const float* __restrict__ bm,
                                 float* __restrict__ out) {
    int t = blockIdx.x * blockDim.x + threadIdx.x;
    if (t >= NN * CCH) return;
    int node = t >> 6, c = t & 63;
    int b = node / NPTS, n = node % NPTS;
    out[((size_t)b * CCH + c) * NPTS + n] = eluf(fo[t] + bm[c]);
}

// ---------------------------------------------------------------------------
extern "C" void kernel_launch(void* const* d_in, const int* in_sizes, int n_in,
                              void* d_out, int out_size, void* d_ws, size_t ws_size,
                              hipStream_t stream) {
    const float* pf  = (const float*)d_in[0];
    const float* pt  = (const float*)d_in[1];
    const float* W1  = (const float*)d_in[2];
    const float* a1s = (const float*)d_in[3];
    const float* a1d = (const float*)d_in[4];
    const float* b1  = (const float*)d_in[5];
    const float* W2  = (const float*)d_in[6];
    const float* a2s = (const float*)d_in[7];
    const float* a2d = (const float*)d_in[8];
    const float* b2  = (const float*)d_in[9];
    const float* Wm  = (const float*)d_in[10];
    const float* bm  = (const float*)d_in[11];
    float* out = (float*)d_out;

    char* ws = (char*)d_ws;
    size_t off = 0;
    auto alloc = [&](size_t bytes) -> void* {
        void* p = ws + off;
        off += (bytes + 255) & ~(size_t)255;
        return p;
    };
    float* xbuf = (float*)alloc((size_t)NN * XPAD * 4);      // padded node table
    float* x2b  = (float*)alloc((size_t)NN * 4);             // squared norms
    float* hbuf = (float*)alloc((size_t)NN * CCH * 4);       // h (and final matmul out)
    float* asrc = (float*)alloc((size_t)NN * HEADS * 4);
    float* adst = (float*)alloc((size_t)NN * HEADS * 4);
    float* fcat = (float*)alloc((size_t)NN * 2 * CCH * 4);   // [f1 | f2]
    float* topd = (float*)alloc((size_t)NN * KNN * 4);
    int*   topi = (int*)  alloc((size_t)NN * KNN * 4);
    float* dchk = (float*)alloc((size_t)NPTS * CHUNK * 4);   // distance column chunk

    const int distBlocks = (NPTS / 32) * (CHUNK / 64) / 4;   // 256 (4 waves/block, 32x64/wave)
    const int mmBlocks   = (NN / 16) / 4;                    // 256 (16x64 per wave)

    // x = [features ; coords], |x|^2
    build_x_kernel<<<(NN + 255) / 256, 256, 0, stream>>>(pf, pt, xbuf, x2b);

    // KNN pass 1 (D = 67, 17 k-steps of 4)
    for (int b = 0; b < BATCH; ++b)
        for (int cb = 0; cb < NPTS; cb += CHUNK) {
            dist_tile_kernel<<<distBlocks, 128, 0, stream>>>(xbuf, x2b, dchk, b, cb, XPAD, 17);
            knn_select_kernel<<<(NPTS + 255) / 256, 256, 0, stream>>>(dchk, topd, topi, b, cb, cb == 0);
        }

    // GAT layer 1
    matmul_wmma_kernel<<<mmBlocks, 128, 0, stream>>>(xbuf, XPAD, W1, DX, hbuf, CCH, 17);
    attn_coeff_kernel<<<(NN * HEADS + 255) / 256, 256, 0, stream>>>(hbuf, a1s, asrc);
    attn_coeff_kernel<<<(NN * HEADS + 255) / 256, 256, 0, stream>>>(hbuf, a1d, adst);
    gat_agg_kernel<<<NN, 64, 0, stream>>>(topi, hbuf, asrc, adst, b1, fcat, 0);

    // KNN pass 2 over f1 (fcat cols 0..63, stride 128, 16 k-steps)
    sqnorm_kernel<<<(NN + 255) / 256, 256, 0, stream>>>(fcat, 2 * CCH, CCH, x2b);
    for (int b = 0; b < BATCH; ++b)
        for (int cb = 0; cb < NPTS; cb += CHUNK) {
            dist_tile_kernel<<<distBlocks, 128, 0, stream>>>(fcat, x2b, dchk, b, cb, 2 * CCH, 16);
            knn_select_kernel<<<(NPTS + 255) / 256, 256, 0, stream>>>(dchk, topd, topi, b, cb, cb == 0);
        }

    // GAT layer 2
    matmul_wmma_kernel<<<mmBlocks, 128, 0, stream>>>(fcat, 2 * CCH, W2, CCH, hbuf, CCH, 16);
    attn_coeff_kernel<<<(NN * HEADS + 255) / 256, 256, 0, stream>>>(hbuf, a2s, asrc);
    attn_coeff_kernel<<<(NN * HEADS + 255) / 256, 256, 0, stream>>>(hbuf, a2d, adst);
    gat_agg_kernel<<<NN, 64, 0, stream>>>(topi, hbuf, asrc, adst, b2, fcat, CCH);

    // Final MLP: [f1|f2] @ Wm, then bias + ELU + transpose
    matmul_wmma_kernel<<<mmBlocks, 128, 0, stream>>>(fcat, 2 * CCH, Wm, 2 * CCH, hbuf, CCH, 32);
    final_out_kernel<<<(NN * CCH + 255) / 256, 256, 0, stream>>>(hbuf, bm, out);
}